// AvgMem_32865089749277
// MI455X (gfx1250) — compile-verified
//
#include <hip/hip_runtime.h>
#include <hip/hip_bf16.h>

// Problem constants (from reference)
#define B        4096      // batch
#define C        10000     // num classes
#define D        1024      // bank dim
#define NSCORE   10000     // score columns
#define CHUNK    1024      // floats per async chunk (256 threads * 4)
#define NCHUNK   10        // ceil(10000/1024)
#define L2M      (-0.15200309344504997f)  // log2(0.9)

// ---------------- CDNA5 async global->LDS helpers ----------------
typedef int v4i __attribute__((ext_vector_type(4)));
typedef __attribute__((address_space(1))) v4i* g_v4i_p;   // global (AS1)
typedef __attribute__((address_space(3))) v4i* l_v4i_p;   // LDS    (AS3)

__device__ __forceinline__ void async_copy16(const float* gsrc, float* ldst) {
#if __has_builtin(__builtin_amdgcn_global_load_async_to_lds_b128)
    __builtin_amdgcn_global_load_async_to_lds_b128(
        (g_v4i_p)gsrc, (l_v4i_p)ldst, /*offset=*/0, /*cpol=*/0);
#else
    unsigned lofs = (unsigned)(unsigned long long)(l_v4i_p)ldst;
    asm volatile("global_load_async_to_lds_b128 %0, %1, off"
                 :: "v"(lofs), "v"(gsrc) : "memory");
#endif
}

template <int N>
__device__ __forceinline__ void wait_async_le() {
#if __has_builtin(__builtin_amdgcn_s_wait_asynccnt)
    __builtin_amdgcn_s_wait_asynccnt(N);
#else
    if (N == 0) asm volatile("s_wait_asynccnt 0" ::: "memory");
    else        asm volatile("s_wait_asynccnt 1" ::: "memory");
#endif
}

// ---------------- K1: argmax per row -> chosen flag ----------------
// Defined FIRST so its device asm appears in the feedback snippet.
// One workgroup (256 threads, 8 waves) per row. Double-buffered async
// global->LDS streaming of 10 x 4KB chunks; per-wave s_wait_asynccnt
// pipelining; first-index tie-break argmax.
__global__ void argmax_chosen_kernel(const float* __restrict__ scores,
                                     const int*   __restrict__ labels,
                                     int*         __restrict__ chosen) {
    __shared__ float buf[2][CHUNK];
    __shared__ float sval[256];
    __shared__ int   sidx[256];

    const int row = blockIdx.x;
    const int tid = threadIdx.x;
    const float* rowp = scores + (size_t)row * NSCORE;

    auto issue = [&](int c) {
        int j0 = c * CHUNK + tid * 4;
        int jc = (j0 > (NSCORE - 4)) ? (NSCORE - 4) : j0;   // clamp last chunk
        async_copy16(rowp + jc, &buf[c & 1][tid * 4]);
    };

    issue(0);
    issue(1);

    float best = -3.402823466e38f;
    int   bidx = 0;

    for (int c = 0; c < NCHUNK; ++c) {
        if (c + 1 < NCHUNK) wait_async_le<1>();   // oldest chunk landed
        else                wait_async_le<0>();
        __syncthreads();

        float4 v = *(const float4*)&buf[c & 1][tid * 4];
        int j0 = c * CHUNK + tid * 4;
        if (j0 < NSCORE) {   // j0 multiple of 4, NSCORE multiple of 4: all-or-none
            if (v.x > best) { best = v.x; bidx = j0;     }
            if (v.y > best) { best = v.y; bidx = j0 + 1; }
            if (v.z > best) { best = v.z; bidx = j0 + 2; }
            if (v.w > best) { best = v.w; bidx = j0 + 3; }
        }
        __syncthreads();                 // all waves done reading buf[c&1]
        if (c + 2 < NCHUNK) issue(c + 2);   // refill same buffer
    }

    sval[tid] = best; sidx[tid] = bidx;
    __syncthreads();
    for (int s = 128; s > 0; s >>= 1) {
        if (tid < s) {
            float ov = sval[tid + s]; int oi = sidx[tid + s];
            if (ov > sval[tid] || (ov == sval[tid] && oi < sidx[tid])) {
                sval[tid] = ov; sidx[tid] = oi;
            }
        }
        __syncthreads();
    }
    if (tid == 0) chosen[row] = (sidx[0] != labels[row]) ? 1 : 0;
}

// ---------------- K0: zero per-class counters ----------------
__global__ void zero_kcnt_kernel(int* __restrict__ kcnt) {
    int g = blockIdx.x * 256 + threadIdx.x;
    if (g < C) kcnt[g] = 0;
}

// ---------------- K2: pairwise c[i], rank r[i], w[i], k[label] ----------------
__global__ void pairwise_kernel(const int* __restrict__ labels,
                                const int* __restrict__ chosen,
                                float* __restrict__ wv,
                                int*   __restrict__ rv,
                                int*   __restrict__ kcnt) {
    __shared__ int pk[B];   // label | (chosen << 16); labels < 2^14
    const int tid = threadIdx.x;
    for (int t = tid; t < B; t += 256) pk[t] = labels[t] | (chosen[t] << 16);
    __syncthreads();

    const int i     = blockIdx.x * 256 + tid;
    const int my    = pk[i];
    const int myLab = my & 0xFFFF;
    const int myCh  = my >> 16;

    int c = 0, r = 0;
    for (int j = 0; j < B; ++j) {
        int pj = pk[j];
        if ((pj >> 16) && ((pj & 0xFFFF) == myLab)) {
            if (j < i) r++; else if (j > i) c++;
        }
    }
    wv[i] = myCh ? 0.1f * exp2f((float)c * L2M) : 0.0f;  // (1-m)*m^c
    rv[i] = r;
    if (myCh) kcnt[myLab] = c + r + 1;   // idempotent across writers: same value
}

// ---------------- K3: exclusive scan of kcnt -> CSR offsets ----------------
__global__ void scan_kernel(const int* __restrict__ kcnt, int* __restrict__ offs) {
    __shared__ int s[1024];
    const int t = threadIdx.x;
    const int base = t * 10;            // 1024 * 10 >= 10000
    int loc[10];
    int sum = 0;
    for (int q = 0; q < 10; ++q) {
        int cls = base + q;
        int v = (cls < C) ? kcnt[cls] : 0;
        loc[q] = sum; sum += v;
    }
    s[t] = sum;
    __syncthreads();
    for (int off = 1; off < 1024; off <<= 1) {   // Hillis-Steele inclusive
        int v = (t >= off) ? s[t - off] : 0;
        __syncthreads();
        s[t] += v;
        __syncthreads();
    }
    int excl = s[t] - sum;
    for (int q = 0; q < 10; ++q) {
        int cls = base + q;
        if (cls < C) offs[cls] = excl + loc[q];
    }
}

// ---------------- K4: deterministic CSR fill ----------------
__global__ void fill_kernel(const int* __restrict__ labels,
                            const int* __restrict__ chosen,
                            const float* __restrict__ wv,
                            const int* __restrict__ rv,
                            const int* __restrict__ offs,
                            int*   __restrict__ idxl,
                            float* __restrict__ wl) {
    int i = blockIdx.x * 256 + threadIdx.x;
    if (i < B && chosen[i]) {
        int slot = offs[labels[i]] + rv[i];   // unique, order = batch order
        idxl[slot] = i;
        wl[slot]   = wv[i];
    }
}

// ---------------- K5: bank/time update (streaming, float4) ----------------
__global__ void bank_kernel(const float* __restrict__ bank,
                            const float* __restrict__ feat,
                            const float* __restrict__ times_in,
                            const int*   __restrict__ kcnt,
                            const int*   __restrict__ offs,
                            const int*   __restrict__ idxl,
                            const float* __restrict__ wl,
                            float* __restrict__ obank,
                            float* __restrict__ otimes) {
    const int l   = blockIdx.x;
    const int tid = threadIdx.x;
    const int cnt = kcnt[l];
    const int off = offs[l];

    const size_t p = (size_t)l * D + tid * 4;
    float4 v = *(const float4*)(bank + p);
    const float scale = exp2f((float)cnt * L2M);   // m^k
    v.x *= scale; v.y *= scale; v.z *= scale; v.w *= scale;

    for (int e = 0; e < cnt; ++e) {                // fixed order -> deterministic
        int   i  = idxl[off + e];
        float wt = wl[off + e];
        float4 f = *(const float4*)(feat + (size_t)i * D + tid * 4);
        v.x += wt * f.x; v.y += wt * f.y; v.z += wt * f.z; v.w += wt * f.w;
    }
    *(float4*)(obank + p) = v;
    if (tid == 0) otimes[l] = times_in[l] + (float)cnt;
}

// ---------------- host launch ----------------
extern "C" void kernel_launch(void* const* d_in, const int* in_sizes, int n_in,
                              void* d_out, int out_size, void* d_ws, size_t ws_size,
                              hipStream_t stream) {
    const float* scores = (const float*)d_in[0];   // [4096, 10000]
    const float* feat   = (const float*)d_in[1];   // [4096, 1024]
    const float* bank   = (const float*)d_in[2];   // [10000, 1024]
    const float* times  = (const float*)d_in[3];   // [10000]
    const int*   labels = (const int*)d_in[4];     // [4096]

    float* obank  = (float*)d_out;                 // [10000*1024]
    float* otimes = obank + (size_t)C * D;         // [10000]

    char* ws = (char*)d_ws;
    int*   chosen = (int*)  (ws + 0);        // 16 KB
    float* wv     = (float*)(ws + 16384);    // 16 KB
    int*   rv     = (int*)  (ws + 32768);    // 16 KB
    int*   kcnt   = (int*)  (ws + 49152);    // 40 KB
    int*   offs   = (int*)  (ws + 90112);    // 40 KB
    int*   idxl   = (int*)  (ws + 131072);   // 16 KB
    float* wl     = (float*)(ws + 147456);   // 16 KB  (total 160 KB)

    zero_kcnt_kernel   <<<(C + 255) / 256, 256, 0, stream>>>(kcnt);
    argmax_chosen_kernel<<<B, 256, 0, stream>>>(scores, labels, chosen);
    pairwise_kernel    <<<B / 256, 256, 0, stream>>>(labels, chosen, wv, rv, kcnt);
    scan_kernel        <<<1, 1024, 0, stream>>>(kcnt, offs);
    fill_kernel        <<<B / 256, 256, 0, stream>>>(labels, chosen, wv, rv, offs, idxl, wl);
    bank_kernel        <<<C, 256, 0, stream>>>(bank, feat, times, kcnt, offs, idxl, wl,
                                               obank, otimes);
}